// KWinnerLayer2D_57088705298666
// MI455X (gfx1250) — compile-verified
//
#include <hip/hip_runtime.h>
#include <hip/hip_bf16.h>
#include <stdint.h>

#define HW    3136      // 56*56
#define KTOP  313       // int(0.1 * 3136)
#define NT    256       // threads per block (8 wave32)

typedef __attribute__((ext_vector_type(4))) unsigned int u32x4;
typedef __attribute__((ext_vector_type(8))) int          i32x8;
typedef __attribute__((ext_vector_type(4))) int          i32x4;

// Issue a 1-D TDM copy of `nelem` f32 from global to LDS (async, TENSORcnt).
// D# per CDNA5 ISA: group0 = {flags, lds_addr,
// global_addr lo, global_addr hi | type=2}; group1 carries data_size,
// tensor/tile dims and stride. Groups 2/3 zero (<=2D tensor).
// amdgpu-toolchain (clang-23) 6-arg builtin form.
__device__ __forceinline__ void tdm_load_row_f32(uint32_t lds_off,
                                                 const float* gptr,
                                                 uint32_t nelem) {
  uint64_t ga = (uint64_t)(uintptr_t)gptr;
  u32x4 g0;
  g0.x = 1u;                                                  // count=1 (valid), user mode
  g0.y = lds_off;                                             // lds_addr (bytes)
  g0.z = (uint32_t)ga;                                        // global_addr[31:0]
  g0.w = (uint32_t)((ga >> 32) & 0x1FFFFFFu) | (2u << 30);    // global_addr[56:32] | type=2
  i32x8 g1;
  g1[0] = (int)(2u << 16);                                    // data_size = 2 -> 4 bytes
  g1[1] = (int)((nelem & 0xFFFFu) << 16);                     // tensor_dim0[15:0]
  g1[2] = (int)((nelem >> 16) | (1u << 16));                  // tensor_dim0[31:16], tensor_dim1=1
  g1[3] = (int)((nelem & 0xFFFFu) << 16);                     // tile_dim0 = nelem
  g1[4] = 0;                                                  // tile_dim1=0, tile_dim2=0
  g1[5] = (int)nelem;                                         // tensor_dim0_stride lo
  g1[6] = 0;
  g1[7] = 0;
  i32x4 z4 = {0, 0, 0, 0};
  i32x8 z8 = {0, 0, 0, 0, 0, 0, 0, 0};
  __builtin_amdgcn_tensor_load_to_lds(g0, g1, z4, z4, z8, 0);
}

// Monotone float->uint key: larger float => larger uint.
__device__ __forceinline__ uint32_t f32_key(float f) {
  uint32_t b = __float_as_uint(f);
  return (b & 0x80000000u) ? ~b : (b | 0x80000000u);
}

__global__ __launch_bounds__(NT)
void kwinner2d_kernel(const float* __restrict__ x,
                      const float* __restrict__ aa,
                      float* __restrict__ out,
                      int C) {
  __shared__ float    xs[HW];     // raw x row (TDM loaded)
  __shared__ uint32_t fk[HW];     // aa row bits (TDM loaded), then xp keys in place
  __shared__ uint32_t hist[256];
  __shared__ uint32_t sel[2];     // [0]=prefix, [1]=k remaining

  const int    row   = blockIdx.x;        // b*C + c
  const int    c     = row - (row / C) * C;
  const int    tid   = threadIdx.x;
  const size_t xbase = (size_t)row * HW;
  const size_t abase = (size_t)c * HW;

  // --- Phase A: async TDM stage of both rows into LDS (wave 0 only) ---
  if (tid == 0) {
    tdm_load_row_f32((uint32_t)(uintptr_t)(void*)xs, x + xbase, HW);
    tdm_load_row_f32((uint32_t)(uintptr_t)(void*)fk, aa + abase, HW);
    __builtin_amdgcn_s_wait_tensorcnt(0);
    sel[1] = KTOP;
  }
  __syncthreads();

  // --- Phase B: compute orderable keys of xp = x * exp(0.1 - aa) in place ---
  for (int i = tid; i < HW; i += NT) {
    float f  = expf(0.1f - __uint_as_float(fk[i]));   // BOOST_FACTOR = 1
    float xp = xs[i] * f;
    fk[i] = f32_key(xp);
  }
  __syncthreads();

  // --- Phase C: 4x8-bit radix select of the KTOP-th largest key ---
  uint32_t prefix = 0, pmask = 0;
  for (int shift = 24; shift >= 0; shift -= 8) {
    hist[tid] = 0;                       // NT == 256 bins
    __syncthreads();
    for (int i = tid; i < HW; i += NT) {
      uint32_t u = fk[i];
      if ((u & pmask) == prefix)
        atomicAdd(&hist[(u >> shift) & 255u], 1u);
    }
    __syncthreads();
    if (tid == 0) {
      uint32_t krem = sel[1];
      int d = 255;
      while (d > 0) {
        uint32_t cnt = hist[d];
        if (krem <= cnt) break;
        krem -= cnt;
        --d;
      }
      sel[0] = prefix | ((uint32_t)d << shift);
      sel[1] = krem;
    }
    __syncthreads();
    prefix = sel[0];
    pmask |= (0xFFu << shift);
  }

  // --- Phase D: masked write-out (non-temporal; output never re-read) ---
  const uint32_t thr = prefix;           // exact key of k-th largest xp
  for (int i = tid; i < HW; i += NT) {
    float v = (fk[i] >= thr) ? xs[i] : 0.0f;
    __builtin_nontemporal_store(v, &out[xbase + i]);
  }
}

extern "C" void kernel_launch(void* const* d_in, const int* in_sizes, int n_in,
                              void* d_out, int out_size, void* d_ws, size_t ws_size,
                              hipStream_t stream) {
  (void)n_in; (void)out_size; (void)d_ws; (void)ws_size;
  const float* x  = (const float*)d_in[0];   // [B,C,H,W] f32
  const float* aa = (const float*)d_in[1];   // [C,H,W]   f32
  float*       out = (float*)d_out;          // [B,C,H,W] f32

  const int chw  = in_sizes[1];              // C*H*W = 128*3136
  const int B    = in_sizes[0] / chw;        // 64
  const int C    = chw / HW;                 // 128
  const int rows = B * C;                    // 8192 blocks, one per (b,c)

  kwinner2d_kernel<<<rows, NT, 0, stream>>>(x, aa, out, C);
}